// LogicLayer_82042465289181
// MI455X (gfx1250) — compile-verified
//
#include <hip/hip_runtime.h>
#include <math.h>
#include <stdint.h>

#define IN_DIM  8192
#define OUT_DIM 8192
#define BATCH   2048
#define ROWS    2          // rows of x staged per block: 2 * 32KB = 64KB LDS
#define TPB     256        // 8 waves (wave32)

// ---------------------------------------------------------------------------
// Kernel 1: fold softmax(weights) of the 16 logic gates into 4 coefficients.
// out[i,j] = c0 + ca*a + cb*b + cab*a*b   (a,b gathered from x row i)
// ---------------------------------------------------------------------------
__global__ __launch_bounds__(TPB) void logic_coeff_kernel(
    const float* __restrict__ w, float4* __restrict__ coef) {
  int j = blockIdx.x * blockDim.x + threadIdx.x;
  if (j >= OUT_DIM) return;
  const float* wj = w + (size_t)j * 16;
  float v[16];
  float m = -3.402823466e38f;
#pragma unroll
  for (int k = 0; k < 16; ++k) { v[k] = wj[k]; m = fmaxf(m, v[k]); }
  float s = 0.0f;
#pragma unroll
  for (int k = 0; k < 16; ++k) { v[k] = __expf(v[k] - m); s += v[k]; }
  float inv = 1.0f / s;
#pragma unroll
  for (int k = 0; k < 16; ++k) v[k] *= inv;

  float c0  = v[8] + v[9] + v[10] + v[11] + v[12] + v[13] + v[14] + v[15];
  float ca  = v[2] + v[3] + v[6] + v[7] - v[8] - v[9] - v[12] - v[13];
  float cb  = v[4] + v[5] + v[6] + v[7] - v[8] - v[9] - v[10] - v[11];
  float cab = v[1] - v[2] - v[4] - 2.0f * v[6] - v[7]
            + v[8] + 2.0f * v[9] + v[11] + v[13] - v[14];
  coef[j] = make_float4(c0, ca, cb, cab);
}

// ---------------------------------------------------------------------------
// CDNA5 async global->LDS copy (ASYNCcnt path), 16 bytes per lane.
// vdst = per-lane LDS byte address, vaddr = per-lane 64-bit global address.
// ---------------------------------------------------------------------------
__device__ __forceinline__ void async_copy_b128(uint32_t lds_byte_addr,
                                                const float* gaddr) {
  asm volatile("global_load_async_to_lds_b128 %0, %1, off"
               :
               : "v"(lds_byte_addr), "v"(gaddr)
               : "memory");
}

__device__ __forceinline__ void wait_asynccnt0() {
  asm volatile("s_wait_asynccnt 0" ::: "memory");
}

__device__ __forceinline__ float eval_gate(float a, float b, float4 c) {
  // c0 + ca*a + cb*b + cab*(a*b), 3 FMAs + 1 mul
  return fmaf(a * b, c.w, fmaf(b, c.z, fmaf(a, c.y, c.x)));
}

// ---------------------------------------------------------------------------
// Kernel 2: each block owns ROWS rows of x. Async-DMA the rows into LDS,
// then gather a/b from LDS per output column and apply the folded bilinear form.
// ---------------------------------------------------------------------------
__global__ __launch_bounds__(TPB) void logic_gate_kernel(
    const float* __restrict__ x, const float4* __restrict__ coef,
    const int* __restrict__ idx_a, const int* __restrict__ idx_b,
    float* __restrict__ out) {
  __shared__ float rowbuf[ROWS * IN_DIM];  // 64 KB

  const int row0 = blockIdx.x * ROWS;

  // ---- Stage ROWS rows of x into LDS via async DMA (streams x once). ----
  {
    const float* src = x + (size_t)row0 * IN_DIM;
    const int nchunks = (ROWS * IN_DIM) / 4;  // float4 chunks
    for (int c = threadIdx.x; c < nchunks; c += TPB) {
      uint32_t lds_addr = (uint32_t)(uintptr_t)(&rowbuf[c * 4]);
      async_copy_b128(lds_addr, src + c * 4);
    }
    wait_asynccnt0();   // drain this wave's ASYNCcnt
    __syncthreads();    // make all waves' LDS writes visible block-wide
  }

  // ---- Compute: 4 consecutive columns per thread per iteration. ----
  for (int jb = threadIdx.x * 4; jb < OUT_DIM; jb += TPB * 4) {
    const int4 ia4 = *reinterpret_cast<const int4*>(idx_a + jb);
    const int4 ib4 = *reinterpret_cast<const int4*>(idx_b + jb);
    const float4 cf0 = coef[jb + 0];
    const float4 cf1 = coef[jb + 1];
    const float4 cf2 = coef[jb + 2];
    const float4 cf3 = coef[jb + 3];

#pragma unroll
    for (int r = 0; r < ROWS; ++r) {
      const float* rb = rowbuf + r * IN_DIM;
      float4 o;
      o.x = eval_gate(rb[ia4.x], rb[ib4.x], cf0);
      o.y = eval_gate(rb[ia4.y], rb[ib4.y], cf1);
      o.z = eval_gate(rb[ia4.z], rb[ib4.z], cf2);
      o.w = eval_gate(rb[ia4.w], rb[ib4.w], cf3);
      *reinterpret_cast<float4*>(out + (size_t)(row0 + r) * OUT_DIM + jb) = o;
    }
  }
}

// ---------------------------------------------------------------------------
// Launcher. Inputs (setup_inputs order): x, weights, idx_a, idx_b.
// ---------------------------------------------------------------------------
extern "C" void kernel_launch(void* const* d_in, const int* in_sizes, int n_in,
                              void* d_out, int out_size, void* d_ws, size_t ws_size,
                              hipStream_t stream) {
  const float* x     = (const float*)d_in[0];
  const float* w     = (const float*)d_in[1];
  const int*   idx_a = (const int*)d_in[2];
  const int*   idx_b = (const int*)d_in[3];
  float*       out   = (float*)d_out;
  float4*      coef  = (float4*)d_ws;  // 8192 * 16B = 128 KB scratch

  logic_coeff_kernel<<<OUT_DIM / TPB, TPB, 0, stream>>>(w, coef);
  logic_gate_kernel<<<BATCH / ROWS, TPB, 0, stream>>>(x, coef, idx_a, idx_b, out);
}